// COUNTYOD_73504070303960
// MI455X (gfx1250) — compile-verified
//
#include <hip/hip_runtime.h>

// Problem constants (match reference)
#define NNODES 50000
#define NEDGES 800000
#define FEAT   96      // H*HN = H*HE = 3*32
#define ODFK   16
#define NSEL   4096

typedef __attribute__((ext_vector_type(2))) float v2f;
typedef __attribute__((ext_vector_type(8))) float v8f;

// ---------------------------------------------------------------------------
// C[M x 96] = act(A[M x K]) @ W[K x 96]     (fp32 WMMA 16x16x4)
// Block: 256 threads = 8 wave32s; each wave owns 16 rows -> block covers 128.
// W staged in LDS in k-pair-interleaved layout so each lane's B fragment
// {W[k][c], W[k+1][c]} is a single aligned b64 LDS load (no packing movs).
// A tile staged row-major per wave; its fragment pair is naturally contiguous.
// Safe for in-place A==C (each block fully loads its rows before storing).
// ---------------------------------------------------------------------------
template <int K, int RELUA>
__global__ void gemm_n96(const float* __restrict__ A, const float* __restrict__ W,
                         float* __restrict__ C, int M)
{
    extern __shared__ float lds[];
    float* sW = lds;                                   // K*96 floats, interleaved
    const int tid  = threadIdx.x;
    const int wave = tid >> 5;
    const int lane = tid & 31;
    float* sA = lds + K * 96 + wave * (16 * K);        // 16 x K per wave

    // Interleave: sW[(k/2)*192 + 2*c + (k&1)] = W[k][c]
    for (int i = tid; i < K * 96; i += 256) {
        int k = i / 96, c = i - k * 96;
        sW[(k >> 1) * 192 + (c << 1) + (k & 1)] = W[i];
    }

    const int rowBase = blockIdx.x * 128 + wave * 16;
    for (int i = lane; i < 16 * K; i += 32) {
        int r = i / K, c = i - r * K;
        int gr = rowBase + r;
        float v = (gr < M) ? A[(long long)gr * K + c] : 0.0f;
        if (RELUA) v = v > 0.0f ? v : 0.0f;
        sA[i] = v;
    }
    __syncthreads();

    // Fragment mapping (ISA 7.12.2, 32-bit A 16x4 / C 16x16):
    //  A: lane%16 = row; lanes>=16 hold K+2..K+3 -> koff = 2*(lane/16)
    //  B (4x16, mirrored): lane%16 = col; lanes>=16 hold K+2..K+3
    const int r    = lane & 15;
    const int koff = (lane >> 4) << 1;

    const v2f* sWp = (const v2f*)sW;   // v2f index: (k0/2)*96 + c
    const v2f* sAp = (const v2f*)sA;   // v2f index: (r*K + k0)/2  (K even)

    v8f acc[6] = {};
    for (int kb = 0; kb < K; kb += 4) {
        const int k0 = kb + koff;                  // even
        v2f a = sAp[(r * K + k0) >> 1];
        const int pb = (k0 >> 1) * 96;
#pragma unroll
        for (int t = 0; t < 6; ++t) {
            v2f b = sWp[pb + t * 16 + r];
            acc[t] = __builtin_amdgcn_wmma_f32_16x16x4_f32(
                false, a, false, b, (short)0, acc[t], false, false);
        }
    }

    // C layout: lanes 0-15 hold M=v, lanes 16-31 hold M=v+8; col = lane%16
    const int rsel = 8 * (lane >> 4);
#pragma unroll
    for (int t = 0; t < 6; ++t) {
#pragma unroll
        for (int v = 0; v < 8; ++v) {
            int row = rowBase + rsel + v;
            if (row < M) C[(long long)row * 96 + t * 16 + r] = acc[t][v];
        }
    }
}

// ---------------------------------------------------------------------------
// Per-edge combine: f = leaky_relu(efW + ni[src] + nj[dst] + bias)
// writes f back, computes per-head attention logit, atomic segment-max (key-
// transformed uint so unsigned atomicMax == float max).
// blockDim = 96 (3 wave32s, one head per wave), grid = E.
// ---------------------------------------------------------------------------
__global__ void edge_combine(const float* __restrict__ tmpF,
                             const float* __restrict__ ni,
                             const float* __restrict__ nj,
                             const float* __restrict__ bias,
                             const float* __restrict__ attn,
                             const int* __restrict__ src,
                             const int* __restrict__ dst,
                             float* __restrict__ fout,
                             float* __restrict__ elog,
                             unsigned int* __restrict__ mkey)
{
    const int e = blockIdx.x;
    const int t = threadIdx.x;            // 0..95
    const int s = src[e], d = dst[e];
    float v = tmpF[(long long)e * 96 + t] + ni[(long long)s * 96 + t]
            + nj[(long long)d * 96 + t] + bias[t];
    v = (v > 0.0f) ? v : 0.01f * v;       // leaky_relu(0.01)
    fout[(long long)e * 96 + t] = v;

    float p = v * attn[t];
#pragma unroll
    for (int off = 16; off > 0; off >>= 1) p += __shfl_xor(p, off, 32);

    if ((t & 31) == 0) {
        int h = t >> 5;
        elog[e * 3 + h] = p;
        unsigned int b = __float_as_uint(p);
        unsigned int k = (b & 0x80000000u) ? ~b : (b | 0x80000000u);
        atomicMax(&mkey[d * 3 + h], k);
    }
}

// ex = exp(e - max[dst]); segment-sum via atomicAdd
__global__ void edge_softmax(const float* __restrict__ elog,
                             const int* __restrict__ dst,
                             const unsigned int* __restrict__ mkey,
                             float* __restrict__ exv,
                             float* __restrict__ ssum, int total)
{
    int i = blockIdx.x * blockDim.x + threadIdx.x;
    if (i >= total) return;
    int e = i / 3, h = i - e * 3;
    int d = dst[e];
    unsigned int k = mkey[d * 3 + h];
    unsigned int b = (k & 0x80000000u) ? (k ^ 0x80000000u) : ~k;
    float m  = __uint_as_float(b);
    float ex = __expf(elog[i] - m);
    exv[i] = ex;
    atomicAdd(&ssum[d * 3 + h], ex);
}

// messages: hout[dst] += hs[src] * (ex / ssum[dst]); one thread per (edge,feat)
__global__ void message_pass(const float* __restrict__ hs,
                             const float* __restrict__ exv,
                             const float* __restrict__ ssum,
                             const int* __restrict__ src,
                             const int* __restrict__ dst,
                             float* __restrict__ hout, int total)
{
    int i = blockIdx.x * blockDim.x + threadIdx.x;
    if (i >= total) return;
    int e = i / 96, f = i - e * 96;
    int h = f >> 5;
    int s = src[e], d = dst[e];
    float a = exv[e * 3 + h] / ssum[d * 3 + h];
    atomicAdd(&hout[(long long)d * 96 + f], hs[(long long)s * 96 + f] * a);
}

__global__ void gather_rows(const float* __restrict__ nh,
                            const int* __restrict__ idxs,
                            float* __restrict__ out, int total)
{
    int i = blockIdx.x * blockDim.x + threadIdx.x;
    if (i >= total) return;
    int r = i / 96, f = i - r * 96;
    out[i] = nh[(long long)idxs[r] * 96 + f];
}

// ---------------------------------------------------------------------------
extern "C" void kernel_launch(void* const* d_in, const int* in_sizes, int n_in,
                              void* d_out, int out_size, void* d_ws, size_t ws_size,
                              hipStream_t stream)
{
    const float* ndata   = (const float*)d_in[0];
    const float* odfeats = (const float*)d_in[1];
    const int*   src     = (const int*)d_in[2];
    const int*   dst     = (const int*)d_in[3];
    const int*   idxs    = (const int*)d_in[4];
    const float* Wns0  = (const float*)d_in[5];
    const float* Wni0  = (const float*)d_in[6];
    const float* Wfij0 = (const float*)d_in[7];
    const float* Wnj0  = (const float*)d_in[8];
    const float* attn0 = (const float*)d_in[9];
    const float* bias0 = (const float*)d_in[10];
    const float* WnsS  = (const float*)d_in[11];
    const float* WniS  = (const float*)d_in[12];
    const float* WfijS = (const float*)d_in[13];
    const float* WnjS  = (const float*)d_in[14];
    const float* attnS = (const float*)d_in[15];
    const float* biasS = (const float*)d_in[16];

    // Workspace carve-up (floats)
    float* buf_ef = (float*)d_ws;                       // E x 96 (in-place f / ef)
    float* nfA  = buf_ef + (size_t)NEDGES * 96;         // N x 96
    float* nfB  = nfA + (size_t)NNODES * 96;            // N x 96
    float* ni   = nfB + (size_t)NNODES * 96;            // N x 96
    float* nj   = ni  + (size_t)NNODES * 96;            // N x 96
    float* hs   = nj  + (size_t)NNODES * 96;            // N x 96
    float* elog = hs  + (size_t)NNODES * 96;            // E x 3
    float* exv  = elog + (size_t)NEDGES * 3;            // E x 3
    float* ssum = exv  + (size_t)NEDGES * 3;            // N x 3
    unsigned int* mkey = (unsigned int*)(ssum + (size_t)NNODES * 3); // N x 3

    const int nodeGrid = (NNODES + 127) / 128;
    const int edgeGrid = (NEDGES + 127) / 128;
    const size_t lds96 = (size_t)(96 * 96 + 8 * 16 * 96) * sizeof(float);
    const size_t lds16 = (size_t)(16 * 96 + 8 * 16 * 16) * sizeof(float);

    for (int l = 0; l < 3; ++l) {
        const float* nfin = (l == 0) ? ndata : (l == 1 ? nfA : nfB);
        float* hout = (l == 1) ? nfB : nfA;   // l0->nfA, l1->nfB, l2->nfA
        const float* Wni_l  = (l == 0) ? Wni0  : WniS  + (size_t)(l - 1) * 96 * 96;
        const float* Wnj_l  = (l == 0) ? Wnj0  : WnjS  + (size_t)(l - 1) * 96 * 96;
        const float* Wns_l  = (l == 0) ? Wns0  : WnsS  + (size_t)(l - 1) * 96 * 96;
        const float* Wfij_l = (l == 0) ? Wfij0 : WfijS + (size_t)(l - 1) * 96 * 96;
        const float* attn_l = (l == 0) ? attn0 : attnS + (size_t)(l - 1) * 96;
        const float* bias_l = (l == 0) ? bias0 : biasS + (size_t)(l - 1) * 96;

        hipMemsetAsync(hout, 0, (size_t)NNODES * 96 * sizeof(float), stream);
        hipMemsetAsync(ssum, 0, (size_t)NNODES * 3 * sizeof(float), stream);
        hipMemsetAsync(mkey, 0, (size_t)NNODES * 3 * sizeof(unsigned int), stream);

        if (l == 0) {
            gemm_n96<96, 0><<<nodeGrid, 256, lds96, stream>>>(nfin, Wni_l, ni, NNODES);
            gemm_n96<96, 0><<<nodeGrid, 256, lds96, stream>>>(nfin, Wnj_l, nj, NNODES);
            gemm_n96<96, 0><<<nodeGrid, 256, lds96, stream>>>(nfin, Wns_l, hs, NNODES);
            gemm_n96<ODFK, 0><<<edgeGrid, 256, lds16, stream>>>(odfeats, Wfij_l, buf_ef, NEDGES);
        } else {
            // inter-layer ReLU folded into the A-tile load
            gemm_n96<96, 1><<<nodeGrid, 256, lds96, stream>>>(nfin, Wni_l, ni, NNODES);
            gemm_n96<96, 1><<<nodeGrid, 256, lds96, stream>>>(nfin, Wnj_l, nj, NNODES);
            gemm_n96<96, 1><<<nodeGrid, 256, lds96, stream>>>(nfin, Wns_l, hs, NNODES);
            gemm_n96<96, 1><<<edgeGrid, 256, lds96, stream>>>(buf_ef, Wfij_l, buf_ef, NEDGES);
        }

        edge_combine<<<NEDGES, 96, 0, stream>>>(buf_ef, ni, nj, bias_l, attn_l,
                                                src, dst, buf_ef, elog, mkey);
        int e3 = NEDGES * 3;
        edge_softmax<<<(e3 + 255) / 256, 256, 0, stream>>>(elog, dst, mkey, exv, ssum, e3);
        int ef = NEDGES * 96;
        message_pass<<<(ef + 255) / 256, 256, 0, stream>>>(hs, exv, ssum, src, dst, hout, ef);
    }

    int tot = NSEL * 96;
    gather_rows<<<(tot + 255) / 256, 256, 0, stream>>>(nfA, idxs, (float*)d_out, tot);
}